// SimMatrixLoss_68289980006459
// MI455X (gfx1250) — compile-verified
//
#include <hip/hip_runtime.h>

// gram[b,i,j] = exp(-sqrt(max(||x_i||^2 + ||x_j||^2 - 2 x_i.x_j, 1e-12)))
// B=8, N=2048, D=256, fp32 in/out.
//
// Output-write bound (134 MB -> ~5.8us @ 23.3 TB/s). fp32 precision required by
// the exp(-dist) epilogue, so use native V_WMMA_F32_16X16X4_F32 (exact fp32).
// Norm accumulation is done as v2f vector FMAs so v_pk_fma_f32 sources the
// WMMA fragment register pairs directly (no v_mov shuffling).

typedef float v2f __attribute__((ext_vector_type(2)));
typedef float v8f __attribute__((ext_vector_type(8)));

#define NB   8
#define NN   2048
#define DD   256
#define KC   64            // K chunk staged in LDS per iteration
#define LSTR (KC + 4)      // 68-float row stride -> bank = 4*m + c (conflict-free)
#define TILE 64            // 64x64 output tile per 128-thread workgroup

__device__ __forceinline__ float gram_elem(float ni, float nj, float dot) {
    float sq = ni + nj - 2.0f * dot;
    sq = fmaxf(sq, 1e-12f);
    return __expf(-sqrtf(sq));
}

extern "C" __global__ void __launch_bounds__(128)
simmatrix_gram_wmma(const float* __restrict__ x, float* __restrict__ out) {
    __shared__ float ldsA[TILE * LSTR];   // rows iBase..iBase+63, K chunk
    __shared__ float ldsB[TILE * LSTR];   // rows jBase..jBase+63, K chunk
    __shared__ float nI[TILE];            // ||x_i||^2 for tile rows
    __shared__ float nJ[TILE];            // ||x_j||^2 for tile cols

    const int b     = blockIdx.z;
    const int iBase = blockIdx.y * TILE;
    const int jBase = blockIdx.x * TILE;
    const int tid   = threadIdx.x;
    const int wave  = tid >> 5;
    const int lane  = tid & 31;
    const int m     = lane & 15;     // row/col within 16x16 fragment
    const int half  = lane >> 4;     // K-half select for A/B fragments
    const int rOff  = (wave >> 1) << 5;   // 0 or 32: wave's row quadrant
    const int cOff  = (wave &  1) << 5;   // 0 or 32: wave's col quadrant

    const float* xb = x + (size_t)b * NN * DD;

    v8f acc00 = {0.f,0.f,0.f,0.f,0.f,0.f,0.f,0.f};
    v8f acc01 = acc00, acc10 = acc00, acc11 = acc00;
    v2f sqA0 = {0.f, 0.f}, sqA1 = sqA0, sqB0 = sqA0, sqB1 = sqA0;

    for (int kt = 0; kt < DD; kt += KC) {
        __syncthreads();   // previous chunk's compute done before overwrite
        // Stage 64 rows x 64 cols of A-rows and B-rows into LDS.
        // 1024 float4 per tile, 128 threads -> 8 fully-coalesced iterations.
        #pragma unroll
        for (int it = 0; it < 8; ++it) {
            int idx = tid + (it << 7);       // 0..1023
            int row = idx >> 4;              // 16 float4 per row
            int c4  = idx & 15;
            const float4 va = *(const float4*)(xb + (size_t)(iBase + row) * DD + kt + (c4 << 2));
            const float4 vb = *(const float4*)(xb + (size_t)(jBase + row) * DD + kt + (c4 << 2));
            *(float4*)(&ldsA[row * LSTR + (c4 << 2)]) = va;
            *(float4*)(&ldsB[row * LSTR + (c4 << 2)]) = vb;
        }
        __syncthreads();

        // A 16x4 f32 fragment: lane m (half h) holds row m, K = k+2h..k+2h+1.
        // B 4x16 fragment with B = X^T: identical per-lane addressing.
        const float* pa0 = &ldsA[(rOff +      m) * LSTR + (half << 1)];
        const float* pa1 = &ldsA[(rOff + 16 + m) * LSTR + (half << 1)];
        const float* pb0 = &ldsB[(cOff +      m) * LSTR + (half << 1)];
        const float* pb1 = &ldsB[(cOff + 16 + m) * LSTR + (half << 1)];

        #pragma unroll 4
        for (int k = 0; k < KC; k += 4) {
            v2f a0 = *(const v2f*)(pa0 + k);
            v2f a1 = *(const v2f*)(pa1 + k);
            v2f b0 = *(const v2f*)(pb0 + k);
            v2f b1 = *(const v2f*)(pb1 + k);
            acc00 = __builtin_amdgcn_wmma_f32_16x16x4_f32(false, a0, false, b0, (short)0, acc00, false, false);
            acc01 = __builtin_amdgcn_wmma_f32_16x16x4_f32(false, a0, false, b1, (short)0, acc01, false, false);
            acc10 = __builtin_amdgcn_wmma_f32_16x16x4_f32(false, a1, false, b0, (short)0, acc10, false, false);
            acc11 = __builtin_amdgcn_wmma_f32_16x16x4_f32(false, a1, false, b1, (short)0, acc11, false, false);
            // inline squared-norm accumulation: one v_pk_fma_f32 per fragment,
            // sourcing the fragment's register pair directly.
            sqA0 = a0 * a0 + sqA0;
            sqA1 = a1 * a1 + sqA1;
            sqB0 = b0 * b0 + sqB0;
            sqB1 = b1 * b1 + sqB1;
        }
    }

    // Horizontal reduce the v2f accumulators, then merge the two K-halves
    // (lanes m and m+16 hold complementary K subsets of the same row/col).
    float sA0 = sqA0.x + sqA0.y;
    float sA1 = sqA1.x + sqA1.y;
    float sB0 = sqB0.x + sqB0.y;
    float sB1 = sqB1.x + sqB1.y;
    sA0 += __shfl_xor(sA0, 16, 32);
    sA1 += __shfl_xor(sA1, 16, 32);
    sB0 += __shfl_xor(sB0, 16, 32);
    sB1 += __shfl_xor(sB1, 16, 32);

    if (half == 0) {
        if ((wave & 1) == 0) {            // waves 0,2 publish row norms
            nI[rOff +      m] = sA0;
            nI[rOff + 16 + m] = sA1;
        }
        if ((wave >> 1) == 0) {           // waves 0,1 publish col norms
            nJ[cOff +      m] = sB0;
            nJ[cOff + 16 + m] = sB1;
        }
    }
    __syncthreads();

    // Epilogue. C/D layout: VGPR v -> row v + 8*half, lane -> col m.
    const float njA = nJ[cOff +      m];
    const float njB = nJ[cOff + 16 + m];
    float* ob = out + (size_t)b * NN * NN;
    #pragma unroll
    for (int v = 0; v < 8; ++v) {
        const int   r0  = rOff + v + (half << 3);
        const int   r1  = r0 + 16;
        const float ni0 = nI[r0];
        const float ni1 = nI[r1];
        const size_t o0 = (size_t)(iBase + r0) * NN;
        const size_t o1 = (size_t)(iBase + r1) * NN;
        const int    j0 = jBase + cOff + m;
        const int    j1 = j0 + 16;
        ob[o0 + j0] = gram_elem(ni0, njA, acc00[v]);
        ob[o0 + j1] = gram_elem(ni0, njB, acc01[v]);
        ob[o1 + j0] = gram_elem(ni1, njA, acc10[v]);
        ob[o1 + j1] = gram_elem(ni1, njB, acc11[v]);
    }
}

extern "C" void kernel_launch(void* const* d_in, const int* in_sizes, int n_in,
                              void* d_out, int out_size, void* d_ws, size_t ws_size,
                              hipStream_t stream) {
    (void)in_sizes; (void)n_in; (void)out_size; (void)d_ws; (void)ws_size;
    const float* x = (const float*)d_in[0];
    float* out = (float*)d_out;
    dim3 grid(NN / TILE, NN / TILE, NB);   // (32, 32, 8)
    simmatrix_gram_wmma<<<grid, 128, 0, stream>>>(x, out);
}